// MixModule_64527588655455
// MI455X (gfx1250) — compile-verified
//
#include <hip/hip_runtime.h>

// CDNA5 / gfx1250, wave32. Uses V_WMMA_F32_16X16X4_F32 (fp32 matrix pipe),
// which keeps exact fp32 precision; the problem is HBM-bandwidth bound
// (~134 MB moved for ~4.3 GFLOP after folding the one-hot op dimension).

typedef __attribute__((ext_vector_type(2))) float v2f;
typedef __attribute__((ext_vector_type(8))) float v8f;

#define DDIM 128
#define NOPS 4

// ---------------------------------------------------------------------------
// Fold the op dimension: W_eff[f,d] = sum_o w[o]*W[o,f,d]; b_eff[f] = sum_o w[o]*b[o,f]
// ---------------------------------------------------------------------------
__global__ void mix_prep_kernel(const float* __restrict__ W,
                                const float* __restrict__ b,
                                const float* __restrict__ wsel,
                                float* __restrict__ Weff,
                                float* __restrict__ beff) {
    int i = blockIdx.x * blockDim.x + threadIdx.x;
    if (i < DDIM * DDIM) {
        float acc = 0.0f;
#pragma unroll
        for (int o = 0; o < NOPS; ++o) acc += wsel[o] * W[o * DDIM * DDIM + i];
        Weff[i] = acc;
    }
    if (i < DDIM) {
        float bb = 0.0f;
#pragma unroll
        for (int o = 0; o < NOPS; ++o) bb += wsel[o] * b[o * DDIM + i];
        beff[i] = bb;
    }
}

// ---------------------------------------------------------------------------
// y[m,f] = sum_d x[m,d] * Weff[f,d] + beff[f]
// Block: 256 threads = 8 waves. Wave w handles rows [row0+16w, row0+16w+16).
// Weff staged in LDS (64KB) once per block, shared by all 8 waves.
//
// WMMA fp32 16x16x4 operand layout (wave32):
//   A (16x4, 2 VGPRs): lane L<16 holds M=L, K={k0,k0+1}; lane L>=16 holds
//                      M=L-16, K={k0+2,k0+3}  -> per-lane v2f global load.
//   B (4x16, 2 VGPRs): lane L<16 holds N=L,  K={k0,k0+1}; lane L>=16 holds
//                      N=L-16, K={k0+2,k0+3}. B[k][n] = Weff[n][k] so a v2f
//                      LDS read of Weff[n0+(L&15)][k0 + 2*(L>>4) + {0,1}].
//   C/D (16x16, 8 VGPRs): VGPR r, lane L -> M = r + 8*(L>=16), N = L&15.
// ---------------------------------------------------------------------------
__global__ __launch_bounds__(256) void mix_gemm_kernel(const float* __restrict__ x,
                                                       const float* __restrict__ Weff,
                                                       const float* __restrict__ beff,
                                                       float* __restrict__ y) {
    __shared__ float lds_w[DDIM * DDIM];  // 64 KB

    const int tid = threadIdx.x;

    // Cooperative 16-byte-granular copy of Weff into LDS.
    {
        const float4* src = (const float4*)Weff;
        float4* dst = (float4*)lds_w;
#pragma unroll
        for (int i = 0; i < (DDIM * DDIM / 4) / 256; ++i)
            dst[tid + i * 256] = src[tid + i * 256];
    }
    __syncthreads();

    const int lane   = tid & 31;
    const int wave   = tid >> 5;
    const int m15    = lane & 15;
    const int colsel = (lane >> 4) << 1;           // 0 for lanes 0-15, 2 for 16-31
    const int row0   = blockIdx.x * 128 + wave * 16;

    const float* xrow = x + (row0 + m15) * DDIM + colsel;

    v8f acc[8] = {};

    for (int k = 0; k < 32; ++k) {                  // K = 128 in steps of 4
        const v2f A = *(const v2f*)(xrow + 4 * k);
#pragma unroll
        for (int n = 0; n < 8; ++n) {               // 8 N-tiles cover all 128 cols
            const v2f Bv = *(const v2f*)(&lds_w[(n * 16 + m15) * DDIM + colsel + 4 * k]);
            acc[n] = __builtin_amdgcn_wmma_f32_16x16x4_f32(
                /*neg_a=*/false, A, /*neg_b=*/false, Bv,
                /*c_mod=*/(short)0, acc[n], /*reuse_a=*/false, /*reuse_b=*/false);
        }
    }

    const int mbase = (lane >> 4) << 3;             // 0 or 8
#pragma unroll
    for (int n = 0; n < 8; ++n) {
        const float bias = beff[n * 16 + m15];
#pragma unroll
        for (int r = 0; r < 8; ++r) {
            y[(row0 + mbase + r) * DDIM + n * 16 + m15] = acc[n][r] + bias;
        }
    }
}

// ---------------------------------------------------------------------------
extern "C" void kernel_launch(void* const* d_in, const int* in_sizes, int n_in,
                              void* d_out, int out_size, void* d_ws, size_t ws_size,
                              hipStream_t stream) {
    const float* x    = (const float*)d_in[0];   // [16, 8192, 128] fp32
    const float* W    = (const float*)d_in[1];   // [4, 128, 128]   fp32
    const float* b    = (const float*)d_in[2];   // [4, 128]        fp32
    const float* wsel = (const float*)d_in[3];   // [4]             fp32 (one-hot)
    float* yout = (float*)d_out;                 // [16, 8192, 128] fp32

    float* Weff = (float*)d_ws;                  // 16384 floats
    float* beff = Weff + DDIM * DDIM;            // 128 floats

    mix_prep_kernel<<<(DDIM * DDIM + 255) / 256, 256, 0, stream>>>(W, b, wsel, Weff, beff);

    const int rows = out_size / DDIM;            // 131072
    mix_gemm_kernel<<<rows / 128, 256, 0, stream>>>(x, Weff, beff, yout);
}